// Model_48653389529485
// MI455X (gfx1250) — compile-verified
//
#include <hip/hip_runtime.h>

// MI455X / gfx1250, wave32. Matrix work via V_WMMA_F32_16X16X4_F32 (fp32-faithful).
// Implicit-GEMM conv; LDS operands stored channel-PAIR interleaved (v2f) so each
// WMMA fragment half is a single aligned ds_load_b64 (no pair-building v_movs).

typedef float v2f __attribute__((ext_vector_type(2)));
typedef float v8f __attribute__((ext_vector_type(8)));

// ---------------- Layer 1: direct 5x5 conv, stride 2, pad 1, cin=3 cout=8, ReLU
__global__ void __launch_bounds__(256) conv1_relu(const float* __restrict__ x,
                                                  const float* __restrict__ k1,
                                                  float* __restrict__ out) {
  __shared__ float sw[600];
  int tid = threadIdx.x;
  for (int e = tid; e < 600; e += 256) sw[e] = k1[e];
  __syncthreads();
  int o  = blockIdx.x * 256 + tid;                // ((b*8+co)*13+py)*13+px
  int px = o % 13;
  int py = (o / 13) % 13;
  int co = (o / 169) % 8;
  int b  = o / (169 * 8);
  float acc = 0.f;
  #pragma unroll
  for (int ci = 0; ci < 3; ++ci) {
    const float* xin = x + ((size_t)(b * 3 + ci)) * 784;
    #pragma unroll
    for (int kh = 0; kh < 5; ++kh) {
      int iy = py * 2 - 1 + kh;
      if (iy < 0 || iy > 27) continue;
      #pragma unroll
      for (int kw = 0; kw < 5; ++kw) {
        int ix = px * 2 - 1 + kw;
        if (ix < 0 || ix > 27) continue;
        // effective weight W[ci][kh][kw][co] == flat[((ci*5+kh)*5+kw)*8+co]
        acc += xin[iy * 28 + ix] * sw[((ci * 5 + kh) * 5 + kw) * 8 + co];
      }
    }
  }
  out[o] = fmaxf(acc, 0.f);
}

// ---------------- Layers 2-5: 3x3 conv s1 p1, cout=16, ReLU; valid 12x12,
// row/col 12 zero (reference loop-undershoot quirk).
// D(16 x 144) = W(16 x 9*CIN) * P(9*CIN x 144); block = image, 9 waves = 9 N-tiles.
template <int CIN>
__global__ void __launch_bounds__(288) conv3x3_wmma_relu(const float* __restrict__ in,
                                                         const float* __restrict__ w,
                                                         float* __restrict__ out) {
  __shared__ v2f lin[(CIN / 2) * 225];            // [cp][sp] -> {ch 2cp, ch 2cp+1}
  __shared__ v2f lw[(CIN / 2) * 9 * 16];          // [cp][r][m] -> {w(2cp), w(2cp+1)}
  float* linf = (float*)lin;
  float* lwf  = (float*)lw;
  int tid = threadIdx.x;
  int b   = blockIdx.x;
  const float* ib = in + (size_t)b * CIN * 169;
  for (int e = tid; e < CIN * 225; e += 288) {
    int ci = e / 225, r = e % 225;
    int yy = r / 15, xx = r % 15;
    float v = 0.f;
    if (yy >= 1 && yy <= 13 && xx >= 1 && xx <= 13)
      v = ib[ci * 169 + (yy - 1) * 13 + (xx - 1)];
    linf[(((ci >> 1) * 225) + r) * 2 + (ci & 1)] = v;
  }
  for (int e = tid; e < CIN * 144; e += 288) {    // w flat index = k*16+m, k=ci*9+r
    int k = e >> 4, m = e & 15;
    int ci = k / 9, r = k % 9;
    lwf[((((ci >> 1) * 9 + r) * 16) + m) * 2 + (ci & 1)] = w[e];
  }
  __syncthreads();

  int lane = tid & 31;
  int wave = tid >> 5;            // N-tile id 0..8
  int m    = lane & 15;
  int hi   = lane >> 4;           // K-slot half: slots {0,1} vs {2,3}
  int p    = wave * 16 + m;       // output pixel 0..143 in 12x12 valid grid
  int py   = p / 12, px = p % 12;
  int spb  = py * 15 + px;
  v8f c = {};
  #pragma unroll
  for (int r = 0; r < 9; ++r) {
    int kh = r / 3, kw = r % 3;
    int sp = spb + kh * 15 + kw;
    #pragma unroll
    for (int cc = 0; cc < CIN / 4; ++cc) {
      int cp0 = cc * 2 + hi;      // this lane's channel pair = K slots {0,1}/{2,3}
      v2f a  = lw[(cp0 * 9 + r) * 16 + m];   // one ds_load_b64
      v2f bb = lin[cp0 * 225 + sp];          // one ds_load_b64
      c = __builtin_amdgcn_wmma_f32_16x16x4_f32(false, a, false, bb,
                                                (short)0, c, false, false);
    }
  }
  float* ob = out + (size_t)b * 16 * 169;
  #pragma unroll
  for (int i = 0; i < 8; ++i) {                   // D: VGPR i -> M = i + 8*hi
    int co = i + hi * 8;
    ob[co * 169 + py * 13 + px] = fmaxf(c[i], 0.f);
  }
  for (int e = tid; e < 400; e += 288) {          // zero row/col 12, all 16 ch
    int ch = e / 25, r2 = e % 25;
    int yy = (r2 < 13) ? 12 : (r2 - 13);
    int xx = (r2 < 13) ? r2 : 12;
    ob[ch * 169 + yy * 13 + xx] = 0.f;
  }
}

// ---------------- Layer 6: 5x5 conv s2 p1, cin=16 cout=32, NO relu.
// Out (B,32,6,6); valid 5x5, row/col 5 zero. K = 400. One M-half per block.
__global__ void __launch_bounds__(128) conv6_wmma(const float* __restrict__ in,
                                                  const float* __restrict__ w,
                                                  float* __restrict__ out) {
  __shared__ v2f lin[8 * 225];                    // 14.4 KB
  __shared__ v2f lw[8 * 25 * 16];                 // 25.6 KB (this M-half)
  float* linf = (float*)lin;
  float* lwf  = (float*)lw;
  int tid = threadIdx.x;
  int b   = blockIdx.x >> 1;
  int mt  = blockIdx.x & 1;
  const float* ib = in + (size_t)b * 16 * 169;
  for (int e = tid; e < 3600; e += 128) {
    int ci = e / 225, r = e % 225;
    int yy = r / 15, xx = r % 15;
    float v = 0.f;
    if (yy >= 1 && yy <= 13 && xx >= 1 && xx <= 13)
      v = ib[ci * 169 + (yy - 1) * 13 + (xx - 1)];
    linf[(((ci >> 1) * 225) + r) * 2 + (ci & 1)] = v;
  }
  for (int e = tid; e < 6400; e += 128) {         // k = ci*25+r; take this mt's 16 cols
    int k = e >> 4, col = e & 15;
    int ci = k / 25, r = k % 25;
    lwf[((((ci >> 1) * 25 + r) * 16) + col) * 2 + (ci & 1)] = w[k * 32 + mt * 16 + col];
  }
  __syncthreads();
  int lane = tid & 31, wave = tid >> 5;
  float* ob = out + (size_t)b * 1152 + mt * 16 * 36;
  if (wave < 2) {                                 // waves 0,1 = two N-tiles
    int m = lane & 15, hi = lane >> 4;
    int praw = wave * 16 + m;                     // valid pixels < 25
    int p  = praw < 25 ? praw : 24;
    int py = p / 5, px = p % 5;
    int spb = py * 30 + px * 2;                   // stride-2 window base (padded plane)
    v8f c = {};
    #pragma unroll
    for (int r = 0; r < 25; ++r) {
      int kh = r / 5, kw = r % 5;
      int sp = spb + kh * 15 + kw;
      #pragma unroll
      for (int cc = 0; cc < 4; ++cc) {
        int cp0 = cc * 2 + hi;
        v2f a  = lw[(cp0 * 25 + r) * 16 + m];
        v2f bb = lin[cp0 * 225 + sp];
        c = __builtin_amdgcn_wmma_f32_16x16x4_f32(false, a, false, bb,
                                                  (short)0, c, false, false);
      }
    }
    if (praw < 25) {
      #pragma unroll
      for (int i = 0; i < 8; ++i) {
        int co = i + hi * 8;                      // local co within this M-half
        ob[co * 36 + py * 6 + px] = c[i];         // no relu
      }
    }
  } else {                                        // waves 2,3: zero row/col 5
    for (int e = tid - 64; e < 176; e += 64) {
      int ch = e / 11, r2 = e % 11;
      int yy = (r2 < 6) ? 5 : (r2 - 6);
      int xx = (r2 < 6) ? r2 : 5;
      ob[ch * 36 + yy * 6 + xx] = 0.f;
    }
  }
}

// ---------------- FC (1152->10) + softmax; one wave per image.
__global__ void __launch_bounds__(256) fc_softmax(const float* __restrict__ flat,
                                                  const float* __restrict__ wfc,
                                                  const float* __restrict__ bfc,
                                                  float* __restrict__ out) {
  __shared__ float lwfc[10 * 1152];               // 46 KB
  int tid = threadIdx.x;
  for (int e = tid; e < 11520; e += 256) lwfc[e] = wfc[e];
  __syncthreads();
  int lane = tid & 31, wave = tid >> 5;
  int b = blockIdx.x * 8 + wave;
  const float* f = flat + (size_t)b * 1152;
  float acc[10];
  #pragma unroll
  for (int cl = 0; cl < 10; ++cl) acc[cl] = 0.f;
  for (int j = lane; j < 1152; j += 32) {
    float v = f[j];
    #pragma unroll
    for (int cl = 0; cl < 10; ++cl) acc[cl] += v * lwfc[cl * 1152 + j];
  }
  #pragma unroll
  for (int cl = 0; cl < 10; ++cl) {
    #pragma unroll
    for (int off = 16; off >= 1; off >>= 1)
      acc[cl] += __shfl_xor(acc[cl], off, 32);
  }
  if (lane == 0) {
    float lg[10], mx = -3.4e38f;
    #pragma unroll
    for (int cl = 0; cl < 10; ++cl) { lg[cl] = acc[cl] + bfc[cl]; mx = fmaxf(mx, lg[cl]); }
    float s = 0.f;
    #pragma unroll
    for (int cl = 0; cl < 10; ++cl) { lg[cl] = __expf(lg[cl] - mx); s += lg[cl]; }
    float inv = 1.f / s;
    #pragma unroll
    for (int cl = 0; cl < 10; ++cl) out[b * 10 + cl] = lg[cl] * inv;
  }
}

extern "C" void kernel_launch(void* const* d_in, const int* in_sizes, int n_in,
                              void* d_out, int out_size, void* d_ws, size_t ws_size,
                              hipStream_t stream) {
  const float* x   = (const float*)d_in[0];
  const float* K1  = (const float*)d_in[1];
  const float* K2  = (const float*)d_in[2];
  const float* K3  = (const float*)d_in[3];
  const float* K4  = (const float*)d_in[4];
  const float* K5  = (const float*)d_in[5];
  const float* K6  = (const float*)d_in[6];
  const float* Wfc = (const float*)d_in[7];
  const float* bfc = (const float*)d_in[8];
  float* out = (float*)d_out;

  const int B = 8192;
  float* ws   = (float*)d_ws;
  float* c1   = ws;                                  // (B,8,13,13)
  float* bufA = c1 + (size_t)B * 8 * 169;            // (B,16,13,13)
  float* bufB = bufA + (size_t)B * 16 * 169;         // (B,16,13,13)
  float* c6   = c1;                                  // alias; conv1 out dead by then
  // total workspace ~221.5 MB

  conv1_relu<<<(B * 8 * 169) / 256, 256, 0, stream>>>(x, K1, c1);
  conv3x3_wmma_relu<8> <<<B, 288, 0, stream>>>(c1,   K2, bufA);
  conv3x3_wmma_relu<16><<<B, 288, 0, stream>>>(bufA, K3, bufB);
  conv3x3_wmma_relu<16><<<B, 288, 0, stream>>>(bufB, K4, bufA);
  conv3x3_wmma_relu<16><<<B, 288, 0, stream>>>(bufA, K5, bufB);
  conv6_wmma<<<2 * B, 128, 0, stream>>>(bufB, K6, c6);
  fc_softmax<<<B / 8, 256, 0, stream>>>(c6, Wfc, bfc, out);
}